// Model_87333864997434
// MI455X (gfx1250) — compile-verified
//
#include <hip/hip_runtime.h>
#include <hip/hip_bf16.h>

// ---------------------------------------------------------------------------
// Ragged prefix-sum gather (speculative-decode KV-slot compaction).
//   out[0:T1]      = concat_b out_cache_loc[b, 0 : accept_length[b]+1]
//   out[T1:T1+T2]  = concat_b out_cache_loc[b, V-to_free_num[b] : V]
// Zero FLOPs, ~64MB gathered reads + ~64MB packed writes -> HBM/L2 bandwidth
// bound (~5.5us floor @ 23.3TB/s; working set fits the 192MB L2). All bulk
// movement rides the gfx1250 async global<->LDS engine (ASYNCcnt), with
// b128 ops on the aligned row-prefix path.
// ---------------------------------------------------------------------------

#define GLOBAL_AS __attribute__((address_space(1)))
#define LDS_AS    __attribute__((address_space(3)))

typedef int v4i_t __attribute__((ext_vector_type(4)));

__device__ static inline void wait_async0() {
#if __has_builtin(__builtin_amdgcn_s_wait_asynccnt)
  __builtin_amdgcn_s_wait_asynccnt(0);
#else
  asm volatile("s_wait_asynccnt 0" ::: "memory");
#endif
}

__device__ static inline void async_ld_b32(const int* g, int* l) {
  __builtin_amdgcn_global_load_async_to_lds_b32((GLOBAL_AS int*)g,
                                                (LDS_AS int*)l, 0, 0);
}
__device__ static inline void async_st_b32(int* g, const int* l) {
  __builtin_amdgcn_global_store_async_from_lds_b32((GLOBAL_AS int*)g,
                                                   (LDS_AS int*)l, 0, 0);
}
__device__ static inline void async_ld_b128(const int* g, int* l) {
  __builtin_amdgcn_global_load_async_to_lds_b128((GLOBAL_AS v4i_t*)g,
                                                 (LDS_AS v4i_t*)l, 0, 0);
}
__device__ static inline void async_st_b128(int* g, const int* l) {
  __builtin_amdgcn_global_store_async_from_lds_b128((GLOBAL_AS v4i_t*)g,
                                                    (LDS_AS v4i_t*)l, 0, 0);
}

// ---------------------------------------------------------------------------
// Kernel 1: dual exclusive scan over bs rows (one 1024-thread block = 32
// wave32s on one WGP). Writes starts1[], starts2[], totals[0] = T1.
// ---------------------------------------------------------------------------
__global__ __launch_bounds__(1024) void scan_kernel(
    const int* __restrict__ accept_length, const int* __restrict__ to_free_num,
    int bs, int* __restrict__ starts1, int* __restrict__ starts2,
    int* __restrict__ totals) {
  __shared__ int sh1[1024];
  __shared__ int sh2[1024];
  const int tid   = threadIdx.x;
  const int chunk = (bs + 1023) >> 10;
  const int beg   = tid * chunk;
  const int end   = (beg + chunk < bs) ? (beg + chunk) : bs;

  int sum1 = 0, sum2 = 0;
  for (int i = beg; i < end; ++i) {
    sum1 += accept_length[i] + 1;
    sum2 += to_free_num[i];
  }
  sh1[tid] = sum1;
  sh2[tid] = sum2;
  __syncthreads();

  // Hillis-Steele inclusive scan over the 1024 partial sums.
  for (int off = 1; off < 1024; off <<= 1) {
    int a = 0, b = 0;
    if (tid >= off) { a = sh1[tid - off]; b = sh2[tid - off]; }
    __syncthreads();
    if (tid >= off) { sh1[tid] += a; sh2[tid] += b; }
    __syncthreads();
  }

  int r1 = (tid == 0) ? 0 : sh1[tid - 1];
  int r2 = (tid == 0) ? 0 : sh2[tid - 1];
  for (int i = beg; i < end; ++i) {
    starts1[i] = r1; r1 += accept_length[i] + 1;
    starts2[i] = r2; r2 += to_free_num[i];
  }
  if (tid == 1023) totals[0] = sh1[1023];  // T1
}

// ---------------------------------------------------------------------------
// Kernel 2: one 256-thread block (8 wave32s) per batch row.
// Fast path (spans fit LDS): issue ALL async loads (b128 on the 16B-aligned
// prefix body, b32 elsewhere) into disjoint LDS halves, one s_wait_asynccnt 0
// + barrier, then issue all async stores (b128 when the packed destination
// happens to be 16B-aligned). S_ENDPGM's implicit wait-idle drains stores.
// ---------------------------------------------------------------------------
#define TILE 1024

// Generic wave-local tiled copy (fallback for spans > TILE): each wave only
// touches its own LDS slots, so s_wait_asynccnt alone orders load->store.
__device__ static inline void copy_span(const int* __restrict__ src,
                                        int* __restrict__ dst, int len,
                                        int tid, int* buf) {
  for (int base = 0; base < len; base += TILE) {
    const int tile = (len - base < TILE) ? (len - base) : TILE;
    for (int i = tid; i < tile; i += 256) async_ld_b32(src + base + i, &buf[i]);
    wait_async0();
    for (int i = tid; i < tile; i += 256) async_st_b32(dst + base + i, &buf[i]);
    wait_async0();  // LDS slots reusable for next tile
  }
}

__global__ __launch_bounds__(256) void gather_kernel(
    const int* __restrict__ out_cache_loc,
    const int* __restrict__ accept_length, const int* __restrict__ to_free_num,
    const int* __restrict__ starts1, const int* __restrict__ starts2,
    const int* __restrict__ totals, int V, int* __restrict__ dst) {
  __shared__ __align__(16) int buf1[TILE];
  __shared__ __align__(16) int buf2[TILE];
  const int b    = blockIdx.x;
  const int tid  = threadIdx.x;
  const int len1 = accept_length[b] + 1;  // prefix length (1..V)
  const int n2   = to_free_num[b];        // suffix length (0..V-1)
  const int s1   = starts1[b];
  const int T1   = totals[0];

  const int* __restrict__ src1 = out_cache_loc + (long long)b * V;
  const int* __restrict__ src2 = src1 + (V - n2);
  int* __restrict__ dst1 = dst + s1;
  int* __restrict__ dst2 = dst + T1 + starts2[b];

  if (len1 <= TILE && n2 <= TILE) {
    // ---- load phase: everything in flight at once ----
    const int nvec = len1 >> 2;  // 16B-aligned b128 groups of the prefix
    for (int i = tid; i < nvec; i += 256)
      async_ld_b128(src1 + 4 * i, &buf1[4 * i]);
    if (tid < (len1 & 3))  // 0..3 tail elements
      async_ld_b32(src1 + (nvec << 2) + tid, &buf1[(nvec << 2) + tid]);
    for (int i = tid; i < n2; i += 256)  // suffix: arbitrary alignment
      async_ld_b32(src2 + i, &buf2[i]);

    wait_async0();    // my async LDS writes are done
    __syncthreads();  // -> everyone's are visible

    // ---- store phase ----
    if ((s1 & 3) == 0) {  // packed dest 16B-aligned: b128 stores
      for (int i = tid; i < nvec; i += 256)
        async_st_b128(dst1 + 4 * i, &buf1[4 * i]);
      if (tid < (len1 & 3))
        async_st_b32(dst1 + (nvec << 2) + tid, &buf1[(nvec << 2) + tid]);
    } else {
      for (int i = tid; i < len1; i += 256) async_st_b32(dst1 + i, &buf1[i]);
    }
    for (int i = tid; i < n2; i += 256) async_st_b32(dst2 + i, &buf2[i]);
    // stores drained by S_ENDPGM's implicit wait-idle
  } else {
    copy_span(src1, dst1, len1, tid, buf1);
    copy_span(src2, dst2, n2, tid, buf2);
  }
}

// ---------------------------------------------------------------------------
// Launcher. Inputs (setup_inputs order): accept_length[bs], to_free_num[bs],
// out_cache_loc[bs*V]. d_out = int32 bits: [T1 | T2] concatenated.
// Workspace: starts1[bs] | starts2[bs] | totals[1]  (ints).
// ---------------------------------------------------------------------------
extern "C" void kernel_launch(void* const* d_in, const int* in_sizes, int n_in,
                              void* d_out, int out_size, void* d_ws,
                              size_t ws_size, hipStream_t stream) {
  (void)n_in; (void)out_size; (void)ws_size;
  const int* accept_length = (const int*)d_in[0];
  const int* to_free_num   = (const int*)d_in[1];
  const int* out_cache_loc = (const int*)d_in[2];

  const int bs = in_sizes[0];
  const int V  = in_sizes[2] / bs;

  int* starts1 = (int*)d_ws;
  int* starts2 = starts1 + bs;
  int* totals  = starts2 + bs;

  scan_kernel<<<1, 1024, 0, stream>>>(accept_length, to_free_num, bs, starts1,
                                      starts2, totals);
  gather_kernel<<<bs, 256, 0, stream>>>(out_cache_loc, accept_length,
                                        to_free_num, starts1, starts2, totals,
                                        V, (int*)d_out);
}